// ParallelMLP_56392920596547
// MI455X (gfx1250) — compile-verified
//
#include <hip/hip_runtime.h>
#include <hip/hip_bf16.h>
#include <math.h>
#include <stdint.h>

// ---------------- problem constants (match reference) ----------------
#define NUM_EXPERTS 8
#define TOP_K       2
#define D_MODEL     2048
#define HIDDEN      2048
#define N_TOKENS    4096
#define T_SLOTS     (N_TOKENS * TOP_K)          // 8192 routed slots
#define CAP         1280                        // int(1.25*2*4096/8) rounded to 8

// ---------------- GEMM tiling ----------------
#define BM   128
#define BN   128
#define BK   32
#define BKP  (BK + 2)     // A tile row: 32 floats + 2-float pad (TDM pad codes 4/1)
#define BNP  (BN + 4)     // B tile row: 128 floats + 4-float pad (TDM pad codes 6/3)

typedef __attribute__((ext_vector_type(2))) float        v2f;
typedef __attribute__((ext_vector_type(8))) float        v8f;
typedef __attribute__((ext_vector_type(4))) unsigned int v4u;
typedef __attribute__((ext_vector_type(8))) int          v8i;
typedef __attribute__((ext_vector_type(4))) int          v4i;

// =====================================================================
// Tensor Data Mover: 2D tile (tile_d0 contiguous elems x tile_d1 rows,
// fp32) from global -> LDS, with optional LDS row padding so the DMA
// writes straight into our bank-conflict-padded tiles.
//   pad codes: interval n -> pad after 2^(n+1) DWORDs; amount m -> m+1 DWORDs
// =====================================================================
__device__ static inline void tdm_load_2d(unsigned lds_addr, const void* gptr,
                                          unsigned tile_d0, unsigned tile_d1,
                                          unsigned tensor_d0, unsigned tensor_d1,
                                          unsigned stride0_elems,
                                          unsigned pad_interval, unsigned pad_amount)
{
    const unsigned long long ga = (unsigned long long)(uintptr_t)gptr;
    v4u g0;
    g0[0] = 1u;                                        // count=1, user mode
    g0[1] = lds_addr;                                  // LDS byte address
    g0[2] = (unsigned)(ga & 0xFFFFFFFFu);              // global addr lo
    g0[3] = (unsigned)((ga >> 32) & 0x1FFFFFFu) | (2u << 30);  // addr hi | type=2
    v8i g1;
    g1[0] = (int)((2u << 16)                           // data_size = 4 bytes
                | (1u << 20)                           // pad_enable
                | (pad_interval << 22)
                | (pad_amount << 25));
    g1[1] = (int)((tensor_d0 & 0xFFFFu) << 16);        // abar=0 | dim0 lo
    g1[2] = (int)((tensor_d0 >> 16) | ((tensor_d1 & 0xFFFFu) << 16));
    g1[3] = (int)((tensor_d1 >> 16) | (tile_d0 << 16));
    g1[4] = (int)(tile_d1 & 0xFFFFu);                  // tile_dim2 = 0 (2D)
    g1[5] = (int)stride0_elems;                        // dim0 stride lo
    g1[6] = 0;                                         // stride0 hi | stride1 lo
    g1[7] = 0;                                         // stride1 hi
    const v4i z = {0, 0, 0, 0};
#if defined(__clang_major__) && (__clang_major__ >= 23)
    const v8i z8 = {0, 0, 0, 0, 0, 0, 0, 0};
    __builtin_amdgcn_tensor_load_to_lds(g0, g1, z, z, z8, 0);
#else
    __builtin_amdgcn_tensor_load_to_lds(g0, g1, z, z, 0);
#endif
}

__device__ static inline unsigned lds_off(const void* p)
{   // LDS flat addresses carry the offset in the low 32 bits
    return (unsigned)(uintptr_t)p;
}

// =====================================================================
// Routing: stable rank-within-expert via wave32 ballot prefix popcount.
// =====================================================================
__global__ __launch_bounds__(256) void moe_route(
    const int* __restrict__ eidx, const float* __restrict__ ew,
    int* __restrict__ row_token, float* __restrict__ row_weight,
    int* __restrict__ counts_out)
{
    const int tid = threadIdx.x;
    for (int i = tid; i < NUM_EXPERTS * CAP; i += 256) {
        row_token[i]  = -1;
        row_weight[i] = 0.0f;
    }
    __syncthreads();

    const int wave = tid >> 5;   // 8 waves -> one expert each
    const int lane = tid & 31;
    int base = 0;
    for (int j0 = 0; j0 < T_SLOTS; j0 += 32) {
        const int j = j0 + lane;
        const bool match = (eidx[j] == wave);
        const unsigned mlo = (unsigned)__ballot(match);   // wave32 low bits
        const int rank = base + __popc(mlo & ((1u << lane) - 1u));
        if (match && rank < CAP) {
            const int row   = wave * CAP + rank;
            row_token[row]  = j / TOP_K;
            row_weight[row] = ew[j];
        }
        base += __popc(mlo);
    }
    if (lane == 0) counts_out[wave] = base;   // uncapped histogram, as reference
}

__global__ __launch_bounds__(256) void zero_y(float* __restrict__ p, int n4)
{
    const int i = blockIdx.x * blockDim.x + threadIdx.x;
    if (i < n4) ((float4*)p)[i] = make_float4(0.f, 0.f, 0.f, 0.f);
}

// =====================================================================
// GEMM1: h = silu(xg@w1) * (xg@w3). A rows gathered from x via row_token
// (vector loads); w1/w3 tiles double-buffered via TDM async DMA.
// =====================================================================
__global__ __launch_bounds__(256) void moe_gemm1(
    const float* __restrict__ x,  const float* __restrict__ w1,
    const float* __restrict__ w3, const int* __restrict__ row_token,
    float* __restrict__ h)
{
    __shared__ float sA [BM][BKP];
    __shared__ float sB1[2][BK][BNP];
    __shared__ float sB3[2][BK][BNP];
    __shared__ int   stoken[BM];

    const int e  = blockIdx.z;
    const int bm = blockIdx.y;
    const int bn = blockIdx.x;
    const int tid  = threadIdx.x;
    const int lane = tid & 31, wid = tid >> 5;
    const int wm = wid >> 1, wn = wid & 1;      // wave grid 4x2
    const int lr = lane & 15, lh = lane >> 4;   // WMMA lane decomposition

    const int rowbase = e * CAP + bm * BM;
    if (tid < BM) stoken[tid] = row_token[rowbase + tid];

    const float* b1base = w1 + (size_t)e * D_MODEL * HIDDEN + (size_t)bn * BN;
    const float* b3base = w3 + (size_t)e * D_MODEL * HIDDEN + (size_t)bn * BN;

    // prologue: DMA tile 0 into buffer 0 (one wave issues; EXEC-independent)
    if (wid == 0) {
        tdm_load_2d(lds_off(&sB1[0][0][0]), b1base, BN, BK, HIDDEN, D_MODEL, HIDDEN, 6, 3);
        tdm_load_2d(lds_off(&sB3[0][0][0]), b3base, BN, BK, HIDDEN, D_MODEL, HIDDEN, 6, 3);
    }

    v8f acc1[2][4] = {};
    v8f acc3[2][4] = {};

    const int NT = D_MODEL / BK;   // 64 K-tiles
    for (int t = 0; t < NT; ++t) {
        const int cur = t & 1, nxt = cur ^ 1;
        const int k0  = t * BK;
        __syncthreads();                       // prev compute done: sA, buf[nxt] free
        if (wid == 0 && t + 1 < NT) {          // prefetch next weight tiles
            const size_t ko = (size_t)(k0 + BK) * HIDDEN;
            tdm_load_2d(lds_off(&sB1[nxt][0][0]), b1base + ko, BN, BK, HIDDEN, D_MODEL, HIDDEN, 6, 3);
            tdm_load_2d(lds_off(&sB3[nxt][0][0]), b3base + ko, BN, BK, HIDDEN, D_MODEL, HIDDEN, 6, 3);
        }
        // ---- stage A tile (gathered rows of x) while DMA streams B ----
        {
            const int mr = tid >> 3;           // 0..31
            const int kq = (tid & 7) * 4;      // 0..28
#pragma unroll
            for (int it = 0; it < 4; ++it) {
                const int m = mr + it * 32;
                const int tok = stoken[m];
                float4 v = make_float4(0.f, 0.f, 0.f, 0.f);
                if (tok >= 0)
                    v = *(const float4*)(x + (size_t)tok * D_MODEL + k0 + kq);
                sA[m][kq + 0] = v.x; sA[m][kq + 1] = v.y;
                sA[m][kq + 2] = v.z; sA[m][kq + 3] = v.w;
            }
        }
        if (wid == 0) {                        // current tile's 2 loads retired
            if (t + 1 < NT) __builtin_amdgcn_s_wait_tensorcnt(2);
            else            __builtin_amdgcn_s_wait_tensorcnt(0);
        }
        __syncthreads();
        // ---- compute: 8 k-steps x (2x4 subtiles x 2 matrices) WMMAs ----
#pragma unroll
        for (int kk = 0; kk < BK; kk += 4) {
            v2f a[2];
#pragma unroll
            for (int mi = 0; mi < 2; ++mi) {
                const float* ap = &sA[wm * 32 + mi * 16 + lr][kk + 2 * lh];
                a[mi][0] = ap[0]; a[mi][1] = ap[1];
            }
#pragma unroll
            for (int ni = 0; ni < 4; ++ni) {
                const int col = wn * 64 + ni * 16 + lr;
                v2f b1, b3;
                b1[0] = sB1[cur][kk + 2 * lh][col]; b1[1] = sB1[cur][kk + 2 * lh + 1][col];
                b3[0] = sB3[cur][kk + 2 * lh][col]; b3[1] = sB3[cur][kk + 2 * lh + 1][col];
#pragma unroll
                for (int mi = 0; mi < 2; ++mi) {
                    acc1[mi][ni] = __builtin_amdgcn_wmma_f32_16x16x4_f32(
                        false, a[mi], false, b1, (short)0, acc1[mi][ni], false, false);
                    acc3[mi][ni] = __builtin_amdgcn_wmma_f32_16x16x4_f32(
                        false, a[mi], false, b3, (short)0, acc3[mi][ni], false, false);
                }
            }
        }
    }

    // ---- epilogue: silu(a1)*a3 -> h ----
#pragma unroll
    for (int mi = 0; mi < 2; ++mi) {
#pragma unroll
        for (int ni = 0; ni < 4; ++ni) {
#pragma unroll
            for (int r = 0; r < 8; ++r) {
                const float v1 = acc1[mi][ni][r];
                const float v3 = acc3[mi][ni][r];
                const float hv = (v1 / (1.0f + __expf(-v1))) * v3;
                const int m = wm * 32 + mi * 16 + lh * 8 + r;
                const int n = bn * BN + wn * 64 + ni * 16 + lr;
                h[(size_t)(rowbase + m) * HIDDEN + n] = hv;
            }
        }
    }
}

// =====================================================================
// GEMM2: ye = h @ w2[e]; BOTH tiles are dense -> fully TDM-fed, double
// buffered. Epilogue scales by router weight and scatters with fp32
// atomics (<=2 commutative adds after zero-init => deterministic).
// =====================================================================
__global__ __launch_bounds__(256) void moe_gemm2(
    const float* __restrict__ h, const float* __restrict__ w2,
    const int* __restrict__ row_token, const float* __restrict__ row_weight,
    float* __restrict__ y)
{
    __shared__ float sA[2][BM][BKP];
    __shared__ float sB[2][BK][BNP];
    __shared__ int   stoken[BM];
    __shared__ float sw[BM];

    const int e  = blockIdx.z;
    const int bm = blockIdx.y;
    const int bn = blockIdx.x;
    const int tid  = threadIdx.x;
    const int lane = tid & 31, wid = tid >> 5;
    const int wm = wid >> 1, wn = wid & 1;
    const int lr = lane & 15, lh = lane >> 4;

    const int rowbase = e * CAP + bm * BM;
    if (tid < BM) {
        stoken[tid] = row_token[rowbase + tid];
        sw[tid]     = row_weight[rowbase + tid];
    }

    const float* abase = h  + (size_t)rowbase * HIDDEN;
    const float* bbase = w2 + (size_t)e * HIDDEN * D_MODEL + (size_t)bn * BN;

    if (wid == 0) {
        tdm_load_2d(lds_off(&sA[0][0][0]), abase, BK, BM, HIDDEN, CAP * NUM_EXPERTS, HIDDEN, 4, 1);
        tdm_load_2d(lds_off(&sB[0][0][0]), bbase, BN, BK, D_MODEL, HIDDEN, D_MODEL, 6, 3);
    }

    v8f acc[2][4] = {};

    const int NT = HIDDEN / BK;
    for (int t = 0; t < NT; ++t) {
        const int cur = t & 1, nxt = cur ^ 1;
        const int k0  = t * BK;
        __syncthreads();
        if (wid == 0) {
            if (t + 1 < NT) {
                tdm_load_2d(lds_off(&sA[nxt][0][0]), abase + (k0 + BK),
                            BK, BM, HIDDEN, CAP * NUM_EXPERTS, HIDDEN, 4, 1);
                tdm_load_2d(lds_off(&sB[nxt][0][0]), bbase + (size_t)(k0 + BK) * D_MODEL,
                            BN, BK, D_MODEL, HIDDEN, D_MODEL, 6, 3);
                __builtin_amdgcn_s_wait_tensorcnt(2);
            } else {
                __builtin_amdgcn_s_wait_tensorcnt(0);
            }
        }
        __syncthreads();
#pragma unroll
        for (int kk = 0; kk < BK; kk += 4) {
            v2f a[2];
#pragma unroll
            for (int mi = 0; mi < 2; ++mi) {
                const float* ap = &sA[cur][wm * 32 + mi * 16 + lr][kk + 2 * lh];
                a[mi][0] = ap[0]; a[mi][1] = ap[1];
            }
#pragma unroll
            for (int ni = 0; ni < 4; ++ni) {
                const int col = wn * 64 + ni * 16 + lr;
                v2f b;
                b[0] = sB[cur][kk + 2 * lh][col]; b[1] = sB[cur][kk + 2 * lh + 1][col];
#pragma unroll
                for (int mi = 0; mi < 2; ++mi)
                    acc[mi][ni] = __builtin_amdgcn_wmma_f32_16x16x4_f32(
                        false, a[mi], false, b, (short)0, acc[mi][ni], false, false);
            }
        }
    }

    // ---- epilogue: weighted scatter-add into y ----
#pragma unroll
    for (int mi = 0; mi < 2; ++mi) {
#pragma unroll
        for (int ni = 0; ni < 4; ++ni) {
#pragma unroll
            for (int r = 0; r < 8; ++r) {
                const int m   = wm * 32 + mi * 16 + lh * 8 + r;
                const int tok = stoken[m];
                if (tok >= 0) {
                    const float val = acc[mi][ni][r] * sw[m];
                    const int n = bn * BN + wn * 64 + ni * 16 + lr;
                    atomicAdd(y + (size_t)tok * D_MODEL + n, val);
                }
            }
        }
    }
}

// =====================================================================
// Host launcher (graph-capture safe: only kernel launches on stream)
// =====================================================================
extern "C" void kernel_launch(void* const* d_in, const int* in_sizes, int n_in,
                              void* d_out, int out_size, void* d_ws, size_t ws_size,
                              hipStream_t stream)
{
    const float* x  = (const float*)d_in[0];
    const float* ew = (const float*)d_in[1];
    const int*   ei = (const int*)  d_in[2];
    const float* w1 = (const float*)d_in[3];
    const float* w2 = (const float*)d_in[4];
    const float* w3 = (const float*)d_in[5];

    float* y      = (float*)d_out;
    int*   counts = (int*)d_out + (size_t)N_TOKENS * D_MODEL;  // tuple tail

    char*  ws         = (char*)d_ws;
    int*   row_token  = (int*)ws;                                       // 40 KB
    float* row_weight = (float*)(ws + NUM_EXPERTS * CAP * sizeof(int)); // 40 KB
    float* h          = (float*)(ws + 2 * NUM_EXPERTS * CAP * 4);       // 84 MB

    moe_route<<<1, 256, 0, stream>>>(ei, ew, row_token, row_weight, counts);

    const int n4 = (N_TOKENS * D_MODEL) / 4;
    zero_y<<<(n4 + 255) / 256, 256, 0, stream>>>(y, n4);

    dim3 g1(HIDDEN / BN, CAP / BM, NUM_EXPERTS);   // 16 x 10 x 8
    moe_gemm1<<<g1, 256, 0, stream>>>(x, w1, w3, row_token, h);

    dim3 g2(D_MODEL / BN, CAP / BM, NUM_EXPERTS);
    moe_gemm2<<<g2, 256, 0, stream>>>(h, w2, row_token, row_weight, y);
}